// GCNStage1_ComputeNorm_ONNX_51994874085826
// MI455X (gfx1250) — compile-verified
//
#include <hip/hip_runtime.h>
#include <stdint.h>

#ifndef __has_builtin
#define __has_builtin(x) 0
#endif

#define NODES 100000

typedef int   v4i __attribute__((ext_vector_type(4)));
typedef float v4f __attribute__((ext_vector_type(4)));

// ---- gfx1250 async-to-LDS path (ASYNCcnt), guarded so compile never breaks ----
#if __has_builtin(__builtin_amdgcn_global_load_async_to_lds_b128)
#define HAVE_ASYNC_B128 1
#endif

#if __has_builtin(__builtin_amdgcn_s_wait_asynccnt)
#define WAIT_ASYNC(n) __builtin_amdgcn_s_wait_asynccnt(n)
#else
#define WAIT_ASYNC(n) asm volatile("s_wait_asynccnt %0" ::"i"(n) : "memory")
#endif

// ============================================================================
// Kernel 1: in-degree histogram over col.  6.4M non-returning u32 atomics.
// Async double-buffered streaming of col through per-wave LDS slices:
// one b128 async op per lane per tile = 512 B/wave/tile = 128 edges/wave/tile.
// ============================================================================
__global__ __launch_bounds__(256) void deg_hist_kernel(const int* __restrict__ col,
                                                       unsigned* __restrict__ deg,
                                                       int nedges) {
#if defined(HAVE_ASYNC_B128)
  __shared__ int stage[8 * 2 * 128];  // 8 waves/block x 2 buffers x 128 ints = 8 KB
  const int lane   = threadIdx.x & 31;
  const int wib    = threadIdx.x >> 5;
  const int gwave  = (blockIdx.x * blockDim.x + threadIdx.x) >> 5;
  const int nwave  = (gridDim.x * blockDim.x) >> 5;
  const int ntiles = nedges >> 7;  // 128 edges per wave-tile
  int* wbase = &stage[wib * 256];

  int tile = gwave;
  int cur  = 0;
  if (tile < ntiles) {
    __builtin_amdgcn_global_load_async_to_lds_b128(
        (v4i*)(col + (tile << 7) + lane * 4), (v4i*)&wbase[lane * 4], 0, 0);
  }
  while (tile < ntiles) {
    const int nxt = tile + nwave;
    if (nxt < ntiles) {
      // issue next stage, then wait for the older one (in-order completion)
      __builtin_amdgcn_global_load_async_to_lds_b128(
          (v4i*)(col + (nxt << 7) + lane * 4),
          (v4i*)&wbase[(cur ^ 1) * 128 + lane * 4], 0, 0);
      WAIT_ASYNC(1);
    } else {
      WAIT_ASYNC(0);
    }
    const v4i c = *(const v4i*)&wbase[cur * 128 + lane * 4];  // ds_load_b128
    atomicAdd(&deg[c.x], 1u);
    atomicAdd(&deg[c.y], 1u);
    atomicAdd(&deg[c.z], 1u);
    atomicAdd(&deg[c.w], 1u);
    cur ^= 1;
    tile = nxt;
  }
  // tail (nedges % 128) — dead for E = 6.4M but kept for generality
  for (int e = (ntiles << 7) + blockIdx.x * blockDim.x + threadIdx.x; e < nedges;
       e += gridDim.x * blockDim.x)
    atomicAdd(&deg[col[e]], 1u);
#else
  const int nq = nedges >> 2;
  for (int q = blockIdx.x * blockDim.x + threadIdx.x; q < nq;
       q += gridDim.x * blockDim.x) {
    const v4i c = ((const v4i*)col)[q];  // global_load_b128
    atomicAdd(&deg[c.x], 1u);
    atomicAdd(&deg[c.y], 1u);
    atomicAdd(&deg[c.z], 1u);
    atomicAdd(&deg[c.w], 1u);
  }
  for (int e = (nq << 2) + blockIdx.x * blockDim.x + threadIdx.x; e < nedges;
       e += gridDim.x * blockDim.x)
    atomicAdd(&deg[col[e]], 1u);
#endif
}

// ============================================================================
// Kernel 2: dinv[i] = deg[i] > 0 ? rsqrt(deg[i]) : 0   (100K elements, trivial)
// ============================================================================
__global__ __launch_bounds__(256) void deg_rsqrt_kernel(const unsigned* __restrict__ deg,
                                                        float* __restrict__ dinv,
                                                        int n) {
  const int i = blockIdx.x * blockDim.x + threadIdx.x;
  if (i < n) {
    const unsigned d = deg[i];
    dinv[i] = d ? rsqrtf((float)d) : 0.0f;
  }
}

// ============================================================================
// Kernel 3: out[e] = dinv[row[e]] * dinv[col[e]]  — b128 streams, random
// gathers hit the L2-resident 400 KB table, NT b128 stores for the output.
// ============================================================================
__global__ __launch_bounds__(256) void edge_norm_kernel(const int* __restrict__ row,
                                                        const int* __restrict__ col,
                                                        const float* __restrict__ dinv,
                                                        float* __restrict__ out,
                                                        int nedges) {
  const int nq = nedges >> 2;
  const int q  = blockIdx.x * blockDim.x + threadIdx.x;
  if (q < nq) {
    const v4i r = ((const v4i*)row)[q];
    const v4i c = ((const v4i*)col)[q];
    v4f o;
    o.x = dinv[r.x] * dinv[c.x];
    o.y = dinv[r.y] * dinv[c.y];
    o.z = dinv[r.z] * dinv[c.z];
    o.w = dinv[r.w] * dinv[c.w];
    __builtin_nontemporal_store(o, (v4f*)out + q);
  }
  // tail (nedges % 4) — dead for E = 6.4M but kept for generality
  const int rem = nedges & 3;
  if (q < rem) {
    const int e = (nq << 2) + q;
    out[e] = dinv[row[e]] * dinv[col[e]];
  }
}

extern "C" void kernel_launch(void* const* d_in, const int* in_sizes, int n_in,
                              void* d_out, int out_size, void* d_ws, size_t ws_size,
                              hipStream_t stream) {
  const int E = in_sizes[0] / 2;           // edge_index is [2, E] int32
  const int* row = (const int*)d_in[0];
  const int* col = row + E;

  // workspace: [deg: 100000 u32][pad to 256B][dinv: 100000 f32]  (~800 KB)
  unsigned* deg = (unsigned*)d_ws;
  float* dinv =
      (float*)((char*)d_ws + ((NODES * sizeof(unsigned) + 255) & ~(size_t)255));

  (void)hipMemsetAsync(deg, 0, NODES * sizeof(unsigned), stream);  // capture-safe

  deg_hist_kernel<<<1024, 256, 0, stream>>>(col, deg, E);

  deg_rsqrt_kernel<<<(NODES + 255) / 256, 256, 0, stream>>>(deg, dinv, NODES);

  const int nq = E / 4;
  int blocks = (nq + 255) / 256;
  if (blocks < 1) blocks = 1;
  edge_norm_kernel<<<blocks, 256, 0, stream>>>(row, col, dinv, (float*)d_out, E);
}